// Pos_embedding_39161511805397
// MI455X (gfx1250) — compile-verified
//
#include <hip/hip_runtime.h>

// out[b,t,d] = sqrt(1024) * x[b,t,d] + pe[t,d]
// pe[t,c] = (c even ? sin : cos)( t * 10000^(-2c/1024) )
//
// Memory-bound streaming op: 512 MB traffic -> ~22us floor @ 23.3 TB/s.
// PE is regenerated in registers (1 v_exp_f32 + 3 muls for the 4 divisors,
// v_fract_f32 range reduction, v_sin/v_cos in revolutions).
// Non-temporal b128 loads/stores: both streams are 256 MB, no reuse, keep L2 clean.

typedef __attribute__((ext_vector_type(4))) float v4f;

#define DIM_EMB   1024
#define TSTEP     4096

__global__ __launch_bounds__(256)
void pos_embedding_kernel(const float* __restrict__ x,
                          float* __restrict__ out,
                          int nvec) {
    const int tid = blockIdx.x * blockDim.x + threadIdx.x;
    if (tid >= nvec) return;

    const int base = tid << 2;                 // flat element index (fits in 32b: <2^26)
    const int d    = base & (DIM_EMB - 1);     // column, multiple of 4
    const int t    = (base >> 10) & (TSTEP - 1);

    // Constants:
    //  K2     = -2*log2(10000)/1024            (divisor exponent, base-2)
    //  R      = 10000^(-2/1024)                (per-column divisor ratio)
    //  INV2PI folds radians->revolutions into the divisor (HW sin/cos take revs)
    const float K2     = -0.025952563f;
    const float R      = 0.98217187f;
    const float INV2PI = 0.15915494309189535f;
    const float SCALE  = 32.0f;                // sqrt(1024)

    float inv0 = __builtin_amdgcn_exp2f((float)d * K2) * INV2PI; // v_exp_f32
    float inv1 = inv0 * R;
    float inv2 = inv1 * R;
    float inv3 = inv2 * R;

    const float pos = (float)t;
    // angle in revolutions; fract = 1-instr range reduction (period 1 rev)
    float a0 = __builtin_amdgcn_fractf(pos * inv0);
    float a1 = __builtin_amdgcn_fractf(pos * inv1);
    float a2 = __builtin_amdgcn_fractf(pos * inv2);
    float a3 = __builtin_amdgcn_fractf(pos * inv3);

    float p0 = __builtin_amdgcn_sinf(a0);      // even column -> sin
    float p1 = __builtin_amdgcn_cosf(a1);      // odd  column -> cos
    float p2 = __builtin_amdgcn_sinf(a2);
    float p3 = __builtin_amdgcn_cosf(a3);

    v4f xv = __builtin_nontemporal_load((const v4f*)x + tid);   // global_load_b128 (NT)
    v4f ov;
    ov.x = fmaf(xv.x, SCALE, p0);
    ov.y = fmaf(xv.y, SCALE, p1);
    ov.z = fmaf(xv.z, SCALE, p2);
    ov.w = fmaf(xv.w, SCALE, p3);
    __builtin_nontemporal_store(ov, (v4f*)out + tid);           // global_store_b128 (NT)
}

extern "C" void kernel_launch(void* const* d_in, const int* in_sizes, int n_in,
                              void* d_out, int out_size, void* d_ws, size_t ws_size,
                              hipStream_t stream) {
    const float* x  = (const float*)d_in[0];
    float* out      = (float*)d_out;

    const int n    = in_sizes[0];   // 16*4096*1024 = 67,108,864 (multiple of 4)
    const int nvec = n >> 2;        // 16,777,216 float4s

    const int block = 256;          // 8 wave32 waves
    const int grid  = (nvec + block - 1) / block;   // 65536, exact cover
    pos_embedding_kernel<<<grid, block, 0, stream>>>(x, out, nvec);
}